// C_LIF_80504866996739
// MI455X (gfx1250) — compile-verified
//
#include <hip/hip_runtime.h>
#include <stdint.h>

// LIF spiking-neuron forward scan, CDNA5 (gfx1250).
// x[B,N,T] f32 -> out[B,N,T] f32 (binary spikes).
// Bandwidth-bound (256 MiB traffic, ~11.5us @ 23.3 TB/s floor). Strategy:
// async-to-LDS double-buffered streaming of x (non-temporal: read-once),
// sequential recurrence in registers, non-temporal full-line output stores.

#define B_DIM 64
#define N_DIM 4096
#define T_DIM 128
#define ROWS 256          // rows (b,n pairs) per block == blockDim.x
#define TC 32             // timesteps per chunk
#define NC (T_DIM / TC)   // 4 chunks
#define PAD 36            // floats per LDS row (144B: 16B aligned, 2-way bank conflict max)
#define LOADS_PER_THREAD 8 // (ROWS*TC*4 bytes) / (ROWS threads * 16B)

typedef float v4f __attribute__((ext_vector_type(4)));

__device__ __forceinline__ void async_ld16_nt(unsigned lds_off, const float* gaddr) {
    // per-lane: copies 16 bytes from global address gaddr to LDS offset lds_off, tracked by ASYNCcnt.
    // TH_LOAD_NT: x is consumed exactly once -> don't retain in L2.
    asm volatile("global_load_async_to_lds_b128 %0, %1, off th:TH_LOAD_NT"
                 :: "v"(lds_off), "v"(gaddr) : "memory");
}

__device__ __forceinline__ void wait_async_8() {
#if __has_builtin(__builtin_amdgcn_s_wait_asynccnt)
    __builtin_amdgcn_s_wait_asynccnt(8);
#else
    asm volatile("s_wait_asynccnt 8" ::: "memory");
#endif
}

__device__ __forceinline__ void wait_async_0() {
#if __has_builtin(__builtin_amdgcn_s_wait_asynccnt)
    __builtin_amdgcn_s_wait_asynccnt(0);
#else
    asm volatile("s_wait_asynccnt 0" ::: "memory");
#endif
}

__global__ __launch_bounds__(ROWS) void lif_scan_kernel(
    const float* __restrict__ x,
    const float* __restrict__ decay_m,
    const float* __restrict__ decay_s,
    const float* __restrict__ vth,
    float* __restrict__ out)
{
    __shared__ __align__(16) float smem[2 * ROWS * PAD];

    const int tid = threadIdx.x;
    const long long rowBase = (long long)blockIdx.x * ROWS;
    const long long rowG = rowBase + tid;
    const int n = (int)(rowG & (N_DIM - 1));

    const float dM = decay_m[n];
    const float dS = decay_s[n];
    const float vt = vth[n];

    // LDS byte offset of smem base (generic-pointer low 32 bits == LDS offset)
    const unsigned ldsBase = (unsigned)(uintptr_t)(void*)smem;

    // cooperative-load mapping: lane tid covers 16B chunk (tid & 7) of row (tid >> 3),
    // stepping 32 rows per load instruction -> consecutive lanes hit consecutive 16B
    // of the same global cacheline (full-line transactions).
    const int r0 = tid >> 3;          // 0..31
    const int toff = (tid & 7) * 4;   // float offset within chunk row: 0..28

    // --- prologue: issue chunk 0 into buffer 0 ---
    #pragma unroll
    for (int i = 0; i < LOADS_PER_THREAD; ++i) {
        const int r = i * 32 + r0;
        const unsigned lo = ldsBase + (unsigned)(r * PAD + toff) * 4u;
        const float* g = x + (rowBase + r) * T_DIM + toff;
        async_ld16_nt(lo, g);
    }

    float M = 0.0f, S = 0.0f, E = 0.0f, o = 0.0f;

    for (int c = 0; c < NC; ++c) {
        const unsigned curOff = (c & 1) ? (unsigned)(ROWS * PAD) : 0u;

        if (c + 1 < NC) {
            // issue next chunk into the other buffer
            const unsigned nxtOff = ((c + 1) & 1) ? (unsigned)(ROWS * PAD) * 4u : 0u;
            const int t0n = (c + 1) * TC;
            #pragma unroll
            for (int i = 0; i < LOADS_PER_THREAD; ++i) {
                const int r = i * 32 + r0;
                const unsigned lo = ldsBase + nxtOff + (unsigned)(r * PAD + toff) * 4u;
                const float* g = x + (rowBase + r) * T_DIM + t0n + toff;
                async_ld16_nt(lo, g);
            }
            // async ops complete in order: <=8 outstanding means chunk c has landed
            wait_async_8();
        } else {
            wait_async_0();
        }
        __syncthreads();

        // --- compute: 32 sequential recurrence steps from my LDS row ---
        const float4* myrow4 =
            (const float4*)(smem + curOff + (unsigned)tid * PAD);
        v4f ov[TC / 4];
        #pragma unroll
        for (int i = 0; i < TC / 4; ++i) {
            const float4 xv = myrow4[i];
            float xs[4] = {xv.x, xv.y, xv.z, xv.w};
            float os[4];
            #pragma unroll
            for (int k = 0; k < 4; ++k) {
                const float xi = xs[k];
                M = dM * (M + xi);
                S = dS * (S + xi);
                E = dM * E + o * vt;          // o = previous spike
                const float u = M - S - E - vt;
                o = (u > 0.0f) ? 1.0f : 0.0f;
                os[k] = o;
            }
            ov[i][0] = os[0]; ov[i][1] = os[1]; ov[i][2] = os[2]; ov[i][3] = os[3];
        }

        // one full 128B cacheline per thread per chunk; NT store (never re-read)
        v4f* op = (v4f*)(out + (size_t)rowG * T_DIM + (size_t)c * TC);
        #pragma unroll
        for (int i = 0; i < TC / 4; ++i) {
            __builtin_nontemporal_store(ov[i], op + i);
        }

        // all threads done reading this buffer before it is re-targeted at c+2
        __syncthreads();
    }
}

extern "C" void kernel_launch(void* const* d_in, const int* in_sizes, int n_in,
                              void* d_out, int out_size, void* d_ws, size_t ws_size,
                              hipStream_t stream) {
    const float* x       = (const float*)d_in[0];
    const float* decay_m = (const float*)d_in[1];
    const float* decay_s = (const float*)d_in[2];
    const float* vth     = (const float*)d_in[3];
    float* out = (float*)d_out;

    const long long totalRows = (long long)B_DIM * N_DIM; // 262144
    dim3 grid((unsigned)(totalRows / ROWS));              // 1024 blocks
    dim3 block(ROWS);                                     // 256 threads (8 waves)
    lif_scan_kernel<<<grid, block, 0, stream>>>(x, decay_m, decay_s, vth, out);
}